// FKModel_84636625535448
// MI455X (gfx1250) — compile-verified
//
#include <hip/hip_runtime.h>
#include <math.h>
#include <stdint.h>

#define FK_EPS   1e-8f
#define FK_SMALL 1e-3f
#define FK_J     24
#define WSTRIDE  32   // floats per joint in the uniform table

// Constant-address-space pointer: uniform loads through this lower to s_load_b*
// (SMEM path, KMcnt) instead of per-lane global_load (vector path, LOADcnt).
typedef const __attribute__((address_space(4))) float* cas4_fp;

// Per-joint uniform table:
//  [0..8]   R_org (row major 3x3)
//  [9..11]  t_org
//  [12..20] M1 = R_org @ K(u)
//  [21..29] M2 = M1 @ K(u)  (= R_org @ K(u)^2)
//  [30]     n  = ||u||   (axes are ~unit; matches reference theta = ||ang*u||)
__device__ __attribute__((aligned(16))) float g_fk_tbl[FK_J * WSTRIDE];

__global__ void fk_precompute(const float* __restrict__ T_origins,
                              const float* __restrict__ axes) {
    int j = threadIdx.x;
    if (j >= FK_J) return;
    const float* T = T_origins + j * 16;
    float R[9], t[3];
    #pragma unroll
    for (int r = 0; r < 3; ++r) {
        R[r*3+0] = T[r*4+0];
        R[r*3+1] = T[r*4+1];
        R[r*3+2] = T[r*4+2];
        t[r]     = T[r*4+3];
    }
    float ax = axes[j*3+0], ay = axes[j*3+1], az = axes[j*3+2];
    float n  = sqrtf(ax*ax + ay*ay + az*az);
    // K(u) = [[0,-az,ay],[az,0,-ax],[-ay,ax,0]]
    float K[9] = { 0.f, -az,  ay,
                   az,  0.f, -ax,
                  -ay,  ax,  0.f };
    float M1[9], M2[9];
    #pragma unroll
    for (int r = 0; r < 3; ++r)
        #pragma unroll
        for (int c = 0; c < 3; ++c)
            M1[r*3+c] = R[r*3+0]*K[0*3+c] + R[r*3+1]*K[1*3+c] + R[r*3+2]*K[2*3+c];
    #pragma unroll
    for (int r = 0; r < 3; ++r)
        #pragma unroll
        for (int c = 0; c < 3; ++c)
            M2[r*3+c] = M1[r*3+0]*K[0*3+c] + M1[r*3+1]*K[1*3+c] + M1[r*3+2]*K[2*3+c];

    float* w = g_fk_tbl + j * WSTRIDE;
    #pragma unroll
    for (int i = 0; i < 9; ++i) w[i]      = R[i];
    #pragma unroll
    for (int i = 0; i < 3; ++i) w[9 + i]  = t[i];
    #pragma unroll
    for (int i = 0; i < 9; ++i) w[12 + i] = M1[i];
    #pragma unroll
    for (int i = 0; i < 9; ++i) w[21 + i] = M2[i];
    w[30] = n;
    w[31] = 0.f;
}

__global__ void __launch_bounds__(256) fk_main(const float* __restrict__ angles,
                                               float* __restrict__ out,
                                               int B) {
    int b = blockIdx.x * blockDim.x + threadIdx.x;
    if (b >= B) return;

    const float* arow = angles + (size_t)b * FK_J;
    // gfx1250 global_prefetch_b8 on the thread's 96B angle row
    __builtin_prefetch(arow, 0, 3);

    // Single SMEM-side base pointer for the uniform table (one address
    // materialization; all joint accesses fold into s_load immediate offsets).
    cas4_fp wt = reinterpret_cast<cas4_fp>(reinterpret_cast<uintptr_t>(&g_fk_tbl[0]));

    // 24 angles: 96B per thread, 16B-aligned (b*96) -> 6x b128 loads
    float a[FK_J];
    const float4* ap = (const float4*)arow;
    #pragma unroll
    for (int i = 0; i < FK_J / 4; ++i) {
        float4 v = ap[i];
        a[4*i+0] = v.x; a[4*i+1] = v.y; a[4*i+2] = v.z; a[4*i+3] = v.w;
    }

    float* o = out + (size_t)b * 75;
    o[0] = 0.f; o[1] = 0.f; o[2] = 0.f;   // base position

    float R[9], t[3];

    #pragma unroll
    for (int j = 0; j < FK_J; ++j) {
        cas4_fp w = wt + j * WSTRIDE;     // uniform -> s_load (SGPR broadcast)
        float ang = a[j];
        float n   = w[30];
        float th  = fabsf(ang) * n;
        float s   = __sinf(th);
        float c   = __cosf(th);
        float th2 = th * th;
        float th4 = th2 * th2;
        bool small = th < FK_SMALL;
        // v_rcp_f32 (~1 ulp) instead of the IEEE div expansion (~8 VALU ops each)
        float A  = small ? fmaf(th4, 1.f/120.f, fmaf(th2, -1.f/6.f,  1.0f))
                         : (s * __builtin_amdgcn_rcpf(th + FK_EPS));
        float Bc = small ? fmaf(th4, 1.f/720.f, fmaf(th2, -1.f/24.f, 0.5f))
                         : ((1.f - c) * __builtin_amdgcn_rcpf(th2 + FK_EPS));
        float alpha = A  * ang;
        float beta  = Bc * ang * ang;

        // N = R_org + alpha*M1 + beta*M2   (= R_org @ Rj)
        float N[9];
        #pragma unroll
        for (int i = 0; i < 9; ++i)
            N[i] = fmaf(beta, w[21 + i], fmaf(alpha, w[12 + i], w[i]));

        if (j == 0) {
            // T_parent = I  ->  R = N, t = t_org
            #pragma unroll
            for (int i = 0; i < 9; ++i) R[i] = N[i];
            t[0] = w[9]; t[1] = w[10]; t[2] = w[11];
        } else {
            // t <- R @ t_org + t   (uses old R)
            float t0 = fmaf(R[0], w[9], fmaf(R[1], w[10], fmaf(R[2], w[11], t[0])));
            float t1 = fmaf(R[3], w[9], fmaf(R[4], w[10], fmaf(R[5], w[11], t[1])));
            float t2 = fmaf(R[6], w[9], fmaf(R[7], w[10], fmaf(R[8], w[11], t[2])));
            t[0] = t0; t[1] = t1; t[2] = t2;
            // R <- R @ N
            float Rn[9];
            #pragma unroll
            for (int r = 0; r < 3; ++r)
                #pragma unroll
                for (int c2 = 0; c2 < 3; ++c2)
                    Rn[r*3+c2] = fmaf(R[r*3+0], N[0*3+c2],
                                 fmaf(R[r*3+1], N[1*3+c2],
                                      R[r*3+2] * N[2*3+c2]));
            #pragma unroll
            for (int i = 0; i < 9; ++i) R[i] = Rn[i];
        }

        o[3 + 3*j + 0] = t[0];
        o[3 + 3*j + 1] = t[1];
        o[3 + 3*j + 2] = t[2];
    }
}

extern "C" void kernel_launch(void* const* d_in, const int* in_sizes, int n_in,
                              void* d_out, int out_size, void* d_ws, size_t ws_size,
                              hipStream_t stream) {
    (void)n_in; (void)out_size; (void)d_ws; (void)ws_size;
    const float* joint_angles = (const float*)d_in[0];   // (B, 24) f32
    const float* T_origins    = (const float*)d_in[1];   // (24, 4, 4) f32
    const float* axes         = (const float*)d_in[2];   // (24, 3) f32
    float* out = (float*)d_out;                          // (B, 75) f32

    int B = in_sizes[0] / FK_J;

    fk_precompute<<<1, 32, 0, stream>>>(T_origins, axes);
    fk_main<<<(B + 255) / 256, 256, 0, stream>>>(joint_angles, out, B);
}